// periodic_padding_8048768712770
// MI455X (gfx1250) — compile-verified
//
#include <hip/hip_runtime.h>
#include <cstdint>

// Problem constants (fixed by the reference).
#define BB 32
#define CC 8
#define HH 512
#define WW 512
#define EP 2            // elp == erp == 2
#define HP (HH + 2*EP)  // 516
#define WP (WW + 2*EP)  // 516

// out[b,c,i,j] = x[b,c,(i-2) mod 512, (j-2) mod 512]
// (M_mat / N_mat are one-hot selector matrices; the GEMMs are periodic padding.)
//
// One wave (32 lanes) per input row:
//   1) async-stage the 2KB row into LDS (aligned b128, ASYNCcnt)
//   2) s_wait_asynccnt 0
//   3) write 129 aligned float4 blocks per output row from LDS (shift by 2
//      handled at 4B granularity in LDS), plus duplicated boundary rows.

#define ROWS_PER_BLOCK 8  // 8 waves x 32 lanes = 256 threads, 16KB LDS / block

__device__ __forceinline__ uint32_t lds_offset_u32(const void* p) {
  // Low 32 bits of a flat pointer into LDS == wave-relative LDS byte offset
  // (aperture bits live in [63:32]); this is what VDST of
  // global_load_async_to_lds expects.
  return (uint32_t)(uintptr_t)p;
}

__global__ __launch_bounds__(256) void periodic_pad_async_kernel(
    const float* __restrict__ x, float* __restrict__ out) {
  __shared__ float lds[ROWS_PER_BLOCK * WW];  // 16 KB

  const int tid  = threadIdx.x;
  const int wave = tid >> 5;   // 0..7
  const int lane = tid & 31;   // 0..31

  const int r = blockIdx.x * ROWS_PER_BLOCK + wave;  // global input row id
  if (r >= BB * CC * HH) return;                     // (exact fit; uniform per wave)

  float* ldsRow = lds + wave * WW;

  // ---- Stage input row r (contiguous 512 floats at x + r*WW) into LDS ----
  const uint32_t lds_base  = lds_offset_u32(ldsRow);
  const uint32_t goff_base = (uint32_t)((size_t)r * WW * sizeof(float));
#pragma unroll
  for (int q = 0; q < 4; ++q) {
    const uint32_t byteOff = (uint32_t)((lane + 32 * q) * 16);  // 16B aligned
    const uint32_t ldsAddr = lds_base + byteOff;
    const uint32_t gOff    = goff_base + byteOff;
    // GVS mode: mem_addr = SADDR(64b sgpr pair) + VADDR(u32) ; VDST = LDS addr
    asm volatile("global_load_async_to_lds_b128 %0, %1, %2"
                 :
                 : "v"(ldsAddr), "v"(gOff), "s"(x)
                 : "memory");
  }
  asm volatile("s_wait_asynccnt 0" ::: "memory");

  // ---- Which output rows does this input row feed? ----
  const int img = r / HH;  // flattened (b,c)
  const int i   = r % HH;  // row within image
  const int oi0 = i + EP;  // main (interior) output row
  int oi1 = -1;            // duplicated boundary row, if any
  if      (i == HH - 2) oi1 = 0;
  else if (i == HH - 1) oi1 = 1;
  else if (i == 0)      oi1 = HP - 2;
  else if (i == 1)      oi1 = HP - 1;

  // Boundary column block: output cols {0,1,2,3} and {512,513,514,515} both
  // map to source cols {510, 511, 0, 1}.
  float4 edge;
  edge.x = ldsRow[WW - 2];
  edge.y = ldsRow[WW - 1];
  edge.z = ldsRow[0];
  edge.w = ldsRow[1];

  const size_t outImg = (size_t)img * HP * WP;

#pragma unroll
  for (int pass = 0; pass < 2; ++pass) {
    const int oi = (pass == 0) ? oi0 : oi1;
    if (oi < 0) continue;
    // Row byte offset = (img*HP+oi)*WP*4 ; WP*4 = 2064 = 129*16 -> 16B aligned.
    float4* orow = (float4*)(out + outImg + (size_t)oi * WP);
#pragma unroll
    for (int q = 0; q < 4; ++q) {
      const int k = lane + 32 * q;  // float4 block 0..127
      float4 v;
      if (k == 0) {
        v = edge;
      } else {
        const int s = 4 * k - 2;  // source col of out col 4k (interior shift)
        v.x = ldsRow[s + 0];
        v.y = ldsRow[s + 1];
        v.z = ldsRow[s + 2];
        v.w = ldsRow[s + 3];
      }
      orow[k] = v;
    }
    if (lane == 0) orow[128] = edge;  // tail block j = 512..515
  }
}

extern "C" void kernel_launch(void* const* d_in, const int* in_sizes, int n_in,
                              void* d_out, int out_size, void* d_ws, size_t ws_size,
                              hipStream_t stream) {
  // d_in[0] = x (B,C,H,W) f32 ; d_in[1]=M_mat, d_in[2]=N_mat (selector
  // matrices -- structure is exploited analytically, so they are not read).
  const float* x = (const float*)d_in[0];
  float* out     = (float*)d_out;

  const int nrows = BB * CC * HH;                 // 131072 input rows
  dim3 grid(nrows / ROWS_PER_BLOCK);              // 16384 blocks
  dim3 block(256);                                // 8 waves
  periodic_pad_async_kernel<<<grid, block, 0, stream>>>(x, out);
}